// Generator3DLUT_rand_73057393705706
// MI455X (gfx1250) — compile-verified
//
#include <hip/hip_runtime.h>

// 3D LUT trilinear interpolation (33^3 LUT, NCHW float32 in/out).
// Memory-bound: 192 MB streamed -> ~8.3 us floor at 23.3 TB/s HBM.
// Strategy: repack LUT to float4 AoS (b128 gathers, L2-resident),
// NT temporal hints for the one-touch pixel stream, global_prefetch_b8,
// v_med3_f32 clamps, wave32-sized 256-thread blocks (8 waves).

#define LDIM   33
#define LDIM2  (LDIM * LDIM)        // 1089
#define LDIM3  (LDIM * LDIM * LDIM) // 35937
#define PLANE  (1024 * 1024)        // H*W per channel plane
#define QSHIFT 18                   // log2(PLANE/4): float4-quads per plane
#define QMASK  ((1u << QSHIFT) - 1u)

typedef float v4f __attribute__((ext_vector_type(4)));

// ---------------- helpers ----------------
// clip(v,0,1)*32 == med3(v*32, 0, 32) exactly (power-of-2 scale).
__device__ __forceinline__ void prep1(float v, int& i0, float& f) {
  float c  = __builtin_amdgcn_fmed3f(v * (float)(LDIM - 1), 0.0f, (float)(LDIM - 1));
  float fl = fminf(floorf(c), (float)(LDIM - 2));
  f  = c - fl;
  i0 = (int)fl;
}

__device__ __forceinline__ float lerp1(float a, float b, float t) {
  return __builtin_fmaf(t, b - a, a);   // sub+fma, VOPD-pairable
}

__device__ __forceinline__ v4f lerp4(v4f a, v4f b, float t) {
  v4f r;
  r.x = lerp1(a.x, b.x, t);
  r.y = lerp1(a.y, b.y, t);
  r.z = lerp1(a.z, b.z, t);
  r.w = 0.0f;
  return r;
}

// Returns interpolated {r, g, b, 0} for one pixel (packed-LUT path).
__device__ __forceinline__ v4f trilerp_packed(
    const v4f* __restrict__ lut4, float r, float g, float b) {
  int ri, gi, bi; float rf, gf, bf;
  prep1(r, ri, rf);
  prep1(g, gi, gf);
  prep1(b, bi, bf);
  int base = ri + gi * LDIM + bi * LDIM2;
  v4f c000 = lut4[base];
  v4f c100 = lut4[base + 1];
  v4f c010 = lut4[base + LDIM];
  v4f c110 = lut4[base + LDIM + 1];
  v4f c001 = lut4[base + LDIM2];
  v4f c101 = lut4[base + LDIM2 + 1];
  v4f c011 = lut4[base + LDIM2 + LDIM];
  v4f c111 = lut4[base + LDIM2 + LDIM + 1];
  v4f e00 = lerp4(c000, c100, rf);
  v4f e10 = lerp4(c010, c110, rf);
  v4f e01 = lerp4(c001, c101, rf);
  v4f e11 = lerp4(c011, c111, rf);
  v4f f0 = lerp4(e00, e10, gf);
  v4f f1 = lerp4(e01, e11, gf);
  return lerp4(f0, f1, bf);
}

// ---------------- main kernel FIRST in file (disasm snippet shows it) ------
__global__ __launch_bounds__(256) void lut3d_apply_packed(
    const float* __restrict__ x, const v4f* __restrict__ lut4,
    float* __restrict__ out, unsigned nquads) {
  unsigned tid    = blockIdx.x * 256u + threadIdx.x;
  unsigned stride = gridDim.x * 256u;
  for (unsigned q = tid; q < nquads; q += stride) {
    // Prefetch next grid-stride chunk (gfx1250: global_prefetch_b8).
    unsigned qn = q + stride;
    if (qn < nquads) {
      unsigned n2 = qn >> QSHIFT;
      unsigned p2 = qn & QMASK;
      const char* pr2 = (const char*)(x + (size_t)n2 * 3u * PLANE) + (size_t)p2 * 16u;
      __builtin_prefetch(pr2, 0, 0);
      __builtin_prefetch(pr2 + (size_t)PLANE * 4u, 0, 0);
      __builtin_prefetch(pr2 + (size_t)PLANE * 8u, 0, 0);
    }
    unsigned n = q >> QSHIFT;
    unsigned p = q & QMASK;
    const v4f* pr = (const v4f*)(x + (size_t)n * 3u * PLANE) + p;
    const v4f* pg = (const v4f*)((const char*)pr + (size_t)PLANE * 4u);
    const v4f* pb = (const v4f*)((const char*)pr + (size_t)PLANE * 8u);
    // One-touch stream: non-temporal loads keep WGP$/L2 for the LUT.
    v4f r4 = __builtin_nontemporal_load(pr);
    v4f g4 = __builtin_nontemporal_load(pg);
    v4f b4 = __builtin_nontemporal_load(pb);
    v4f o0 = trilerp_packed(lut4, r4.x, g4.x, b4.x);
    v4f o1 = trilerp_packed(lut4, r4.y, g4.y, b4.y);
    v4f o2 = trilerp_packed(lut4, r4.z, g4.z, b4.z);
    v4f o3 = trilerp_packed(lut4, r4.w, g4.w, b4.w);
    v4f orr, og, ob;
    orr.x = o0.x; orr.y = o1.x; orr.z = o2.x; orr.w = o3.x;
    og.x  = o0.y; og.y  = o1.y; og.z  = o2.y; og.w  = o3.y;
    ob.x  = o0.z; ob.y  = o1.z; ob.z  = o2.z; ob.w  = o3.z;
    v4f* qr = (v4f*)(out + (size_t)n * 3u * PLANE) + p;
    v4f* qg = (v4f*)((char*)qr + (size_t)PLANE * 4u);
    v4f* qb = (v4f*)((char*)qr + (size_t)PLANE * 8u);
    __builtin_nontemporal_store(orr, qr);
    __builtin_nontemporal_store(og, qg);
    __builtin_nontemporal_store(ob, qb);
  }
}

// ---------------- LUT repack: planar [3][33^3] -> interleaved v4f[33^3] ----
__global__ __launch_bounds__(256) void repack_lut_kernel(
    const float* __restrict__ lut, v4f* __restrict__ lut4) {
  int i = blockIdx.x * 256 + threadIdx.x;
  if (i < LDIM3) {
    v4f v;
    v.x = lut[i];
    v.y = lut[LDIM3 + i];
    v.z = lut[2 * LDIM3 + i];
    v.w = 0.0f;
    lut4[i] = v;
  }
}

// ---------------- fallback: planar-LUT path (no workspace needed) ----------
__device__ __forceinline__ v4f trilerp_planar(
    const float* __restrict__ lut, float r, float g, float b) {
  int ri, gi, bi; float rf, gf, bf;
  prep1(r, ri, rf);
  prep1(g, gi, gf);
  prep1(b, bi, bf);
  int base = ri + gi * LDIM + bi * LDIM2;
  float tmp[3];
#pragma unroll
  for (int c = 0; c < 3; ++c) {
    const float* L = lut + c * LDIM3 + base;
    float c000 = L[0],             c100 = L[1];
    float c010 = L[LDIM],          c110 = L[LDIM + 1];
    float c001 = L[LDIM2],         c101 = L[LDIM2 + 1];
    float c011 = L[LDIM2 + LDIM],  c111 = L[LDIM2 + LDIM + 1];
    float e00 = lerp1(c000, c100, rf);
    float e10 = lerp1(c010, c110, rf);
    float e01 = lerp1(c001, c101, rf);
    float e11 = lerp1(c011, c111, rf);
    tmp[c] = lerp1(lerp1(e00, e10, gf), lerp1(e01, e11, gf), bf);
  }
  v4f o;
  o.x = tmp[0]; o.y = tmp[1]; o.z = tmp[2]; o.w = 0.0f;
  return o;
}

__global__ __launch_bounds__(256) void lut3d_apply_planar(
    const float* __restrict__ x, const float* __restrict__ lut,
    float* __restrict__ out, unsigned nquads) {
  unsigned tid    = blockIdx.x * 256u + threadIdx.x;
  unsigned stride = gridDim.x * 256u;
  for (unsigned q = tid; q < nquads; q += stride) {
    unsigned n = q >> QSHIFT;
    unsigned p = q & QMASK;
    const v4f* pr = (const v4f*)(x + (size_t)n * 3u * PLANE) + p;
    const v4f* pg = (const v4f*)((const char*)pr + (size_t)PLANE * 4u);
    const v4f* pb = (const v4f*)((const char*)pr + (size_t)PLANE * 8u);
    v4f r4 = __builtin_nontemporal_load(pr);
    v4f g4 = __builtin_nontemporal_load(pg);
    v4f b4 = __builtin_nontemporal_load(pb);
    v4f o0 = trilerp_planar(lut, r4.x, g4.x, b4.x);
    v4f o1 = trilerp_planar(lut, r4.y, g4.y, b4.y);
    v4f o2 = trilerp_planar(lut, r4.z, g4.z, b4.z);
    v4f o3 = trilerp_planar(lut, r4.w, g4.w, b4.w);
    v4f orr, og, ob;
    orr.x = o0.x; orr.y = o1.x; orr.z = o2.x; orr.w = o3.x;
    og.x  = o0.y; og.y  = o1.y; og.z  = o2.y; og.w  = o3.y;
    ob.x  = o0.z; ob.y  = o1.z; ob.z  = o2.z; ob.w  = o3.z;
    v4f* qr = (v4f*)(out + (size_t)n * 3u * PLANE) + p;
    v4f* qg = (v4f*)((char*)qr + (size_t)PLANE * 4u);
    v4f* qb = (v4f*)((char*)qr + (size_t)PLANE * 8u);
    __builtin_nontemporal_store(orr, qr);
    __builtin_nontemporal_store(og, qg);
    __builtin_nontemporal_store(ob, qb);
  }
}

extern "C" void kernel_launch(void* const* d_in, const int* in_sizes, int n_in,
                              void* d_out, int out_size, void* d_ws, size_t ws_size,
                              hipStream_t stream) {
  const float* x   = (const float*)d_in[0];
  const float* lut = (const float*)d_in[1];
  float* out       = (float*)d_out;

  unsigned nimg   = (unsigned)(in_sizes[0] / (3 * PLANE)); // = 8
  unsigned nquads = nimg * (PLANE / 4);                    // float4 quads over all images

  const int blocks = 2048;  // grid-stride; ~4 iters/thread at reference sizes
  if (ws_size >= (size_t)LDIM3 * sizeof(v4f)) {
    v4f* lut4 = (v4f*)d_ws;
    repack_lut_kernel<<<(LDIM3 + 255) / 256, 256, 0, stream>>>(lut, lut4);
    lut3d_apply_packed<<<blocks, 256, 0, stream>>>(x, lut4, out, nquads);
  } else {
    lut3d_apply_planar<<<blocks, 256, 0, stream>>>(x, lut, out, nquads);
  }
}